// MultiHeadSelfAttention_39711267619220
// MI455X (gfx1250) — compile-verified
//
#include <hip/hip_runtime.h>
#include <hip/hip_bf16.h>
#include <math.h>

typedef __attribute__((ext_vector_type(2))) float v2f;
typedef __attribute__((ext_vector_type(8))) float v8f;

#define EMBED   1024
#define NHEAD   16
#define HDIM    64
#define BATCH   2
#define SEQ     2048
#define QKV3    (3 * EMBED)

// ---------------------------------------------------------------------------
// WMMA helper: D = A(16x4 f32) * B(4x16 f32) + C, wave32.
// A frag: lane l holds row m=l&15; lanes<16 -> K={0,1}, lanes>=16 -> K={2,3}.
// B frag: lane l holds col n=l&15; lanes<16 -> K={0,1}, lanes>=16 -> K={2,3}.
// C/D frag: VGPR r: lanes<16 -> M=r, lanes>=16 -> M=r+8; N = lane&15.
// ---------------------------------------------------------------------------
__device__ __forceinline__ v8f wmma_f32(v2f a, v2f b, v8f c) {
    return __builtin_amdgcn_wmma_f32_16x16x4_f32(false, a, false, b,
                                                 (short)0, c, false, false);
}

// ===========================================================================
// Kernel 1/5: OUT[r][f] = sum_e X[r][e] * W[f][e] + bias[f]   (NT GEMM)
// Block tile 64(M) x 128(N); 256 thr = 8 waves (4 M-subtiles x 2 N-halves),
// each wave owns a 16x64 strip via 4 accumulators. K staged 16-wide in LDS.
// Requires M%64==0, F%128==0, Kdim%16==0.
// ===========================================================================
__global__ __launch_bounds__(256)
void gemm_nt_bias(const float* __restrict__ X, const float* __restrict__ W,
                  const float* __restrict__ bias, float* __restrict__ OUT,
                  int Kdim, int F) {
    __shared__ float As[64][20];    // stride 20 floats = 80B (16B aligned rows)
    __shared__ float Bs[128][20];

    const int tid  = threadIdx.x;
    const int lane = tid & 31;
    const int wave = tid >> 5;
    const int wm   = wave & 3;          // M subtile 0..3
    const int wn   = wave >> 2;         // N half 0..1
    const int m16  = lane & 15;
    const int kh   = (lane >> 4) << 1;  // 0 or 2 (K pair within step)
    const int mBase = blockIdx.x * 64;
    const int nBase = blockIdx.y * 128;

    v8f acc0 = {}, acc1 = {}, acc2 = {}, acc3 = {};

    const int ldRow = tid >> 2;         // 0..63
    const int ldCol = (tid & 3) << 2;   // 0,4,8,12

    for (int kb = 0; kb < Kdim; kb += 16) {
        __syncthreads();
        // A tile: 64 x 16
        *(float4*)&As[ldRow][ldCol] =
            *(const float4*)&X[(size_t)(mBase + ldRow) * Kdim + kb + ldCol];
        // B tile: 128 x 16 (rows of W, K-contiguous)
        *(float4*)&Bs[ldRow][ldCol] =
            *(const float4*)&W[(size_t)(nBase + ldRow) * Kdim + kb + ldCol];
        *(float4*)&Bs[ldRow + 64][ldCol] =
            *(const float4*)&W[(size_t)(nBase + ldRow + 64) * Kdim + kb + ldCol];
        __syncthreads();

        const int nb = wn * 64;
        #pragma unroll
        for (int k0 = 0; k0 < 16; k0 += 4) {
            v2f a;
            a.x = As[wm * 16 + m16][k0 + kh];
            a.y = As[wm * 16 + m16][k0 + kh + 1];
            v2f b0, b1, b2, b3;
            b0.x = Bs[nb +      m16][k0 + kh]; b0.y = Bs[nb +      m16][k0 + kh + 1];
            b1.x = Bs[nb + 16 + m16][k0 + kh]; b1.y = Bs[nb + 16 + m16][k0 + kh + 1];
            b2.x = Bs[nb + 32 + m16][k0 + kh]; b2.y = Bs[nb + 32 + m16][k0 + kh + 1];
            b3.x = Bs[nb + 48 + m16][k0 + kh]; b3.y = Bs[nb + 48 + m16][k0 + kh + 1];
            acc0 = wmma_f32(a, b0, acc0);
            acc1 = wmma_f32(a, b1, acc1);
            acc2 = wmma_f32(a, b2, acc2);
            acc3 = wmma_f32(a, b3, acc3);
        }
    }

    const int rowOff = (lane >> 4) * 8;
    const int c0 = nBase + wn * 64 + m16;
    #pragma unroll
    for (int r = 0; r < 8; ++r) {
        const size_t row = (size_t)(mBase + wm * 16 + rowOff + r) * F;
        OUT[row + c0     ] = acc0[r] + bias[c0     ];
        OUT[row + c0 + 16] = acc1[r] + bias[c0 + 16];
        OUT[row + c0 + 32] = acc2[r] + bias[c0 + 32];
        OUT[row + c0 + 48] = acc3[r] + bias[c0 + 48];
    }
}

// ===========================================================================
// Kernel 2/5: scores[b,h,q,k] = (1/8) * sum_d Q[b,h,q,d] * K[b,h,k,d]
// Q/K live interleaved in qkv [b][s][3E]. Block tile 64(q) x 128(k), D=64.
// ===========================================================================
__global__ __launch_bounds__(256)
void scores_kernel(const float* __restrict__ qkv, float* __restrict__ attn) {
    __shared__ float Qs[64][20];
    __shared__ float Ks[128][20];

    const int tid  = threadIdx.x;
    const int lane = tid & 31;
    const int wave = tid >> 5;
    const int wm   = wave & 3;
    const int wn   = wave >> 2;
    const int m16  = lane & 15;
    const int kh   = (lane >> 4) << 1;

    const int bh = blockIdx.z;
    const int b  = bh >> 4;
    const int h  = bh & 15;
    const int qBase = blockIdx.x * 64;
    const int kBase = blockIdx.y * 128;

    const float* Qp = qkv + (size_t)b * SEQ * QKV3 + (size_t)h * HDIM;
    const float* Kp = Qp + EMBED;

    v8f acc0 = {}, acc1 = {}, acc2 = {}, acc3 = {};

    const int ldRow = tid >> 2;
    const int ldCol = (tid & 3) << 2;

    for (int db = 0; db < HDIM; db += 16) {
        __syncthreads();
        *(float4*)&Qs[ldRow][ldCol] =
            *(const float4*)&Qp[(size_t)(qBase + ldRow) * QKV3 + db + ldCol];
        *(float4*)&Ks[ldRow][ldCol] =
            *(const float4*)&Kp[(size_t)(kBase + ldRow) * QKV3 + db + ldCol];
        *(float4*)&Ks[ldRow + 64][ldCol] =
            *(const float4*)&Kp[(size_t)(kBase + ldRow + 64) * QKV3 + db + ldCol];
        __syncthreads();

        const int nb = wn * 64;
        #pragma unroll
        for (int k0 = 0; k0 < 16; k0 += 4) {
            v2f a;
            a.x = Qs[wm * 16 + m16][k0 + kh];
            a.y = Qs[wm * 16 + m16][k0 + kh + 1];
            v2f b0, b1, b2, b3;
            b0.x = Ks[nb +      m16][k0 + kh]; b0.y = Ks[nb +      m16][k0 + kh + 1];
            b1.x = Ks[nb + 16 + m16][k0 + kh]; b1.y = Ks[nb + 16 + m16][k0 + kh + 1];
            b2.x = Ks[nb + 32 + m16][k0 + kh]; b2.y = Ks[nb + 32 + m16][k0 + kh + 1];
            b3.x = Ks[nb + 48 + m16][k0 + kh]; b3.y = Ks[nb + 48 + m16][k0 + kh + 1];
            acc0 = wmma_f32(a, b0, acc0);
            acc1 = wmma_f32(a, b1, acc1);
            acc2 = wmma_f32(a, b2, acc2);
            acc3 = wmma_f32(a, b3, acc3);
        }
    }

    const float scale = 0.125f;   // 1/sqrt(64)
    const int rowOff = (lane >> 4) * 8;
    const int c0 = kBase + wn * 64 + m16;
    float* Arow = attn + (size_t)bh * SEQ * SEQ;
    #pragma unroll
    for (int r = 0; r < 8; ++r) {
        const size_t row = (size_t)(qBase + wm * 16 + rowOff + r) * SEQ;
        Arow[row + c0     ] = acc0[r] * scale;
        Arow[row + c0 + 16] = acc1[r] * scale;
        Arow[row + c0 + 32] = acc2[r] * scale;
        Arow[row + c0 + 48] = acc3[r] * scale;
    }
}

// ===========================================================================
// Kernel 3/5: in-place row softmax over rows of length SEQ=2048.
// One wave per row, 64 floats/lane held in registers: 1 read + 1 write.
// ===========================================================================
__global__ __launch_bounds__(256)
void softmax_kernel(float* __restrict__ attn) {
    const int lane = threadIdx.x & 31;
    const int wave = threadIdx.x >> 5;
    const size_t row = (size_t)blockIdx.x * 8 + wave;   // < B*H*S
    float* p = attn + row * SEQ;

    float4 v[16];
    float m = -INFINITY;
    #pragma unroll
    for (int i = 0; i < 16; ++i) {
        v[i] = *(const float4*)&p[i * 128 + lane * 4];
        m = fmaxf(m, fmaxf(fmaxf(v[i].x, v[i].y), fmaxf(v[i].z, v[i].w)));
    }
    #pragma unroll
    for (int off = 16; off > 0; off >>= 1)
        m = fmaxf(m, __shfl_xor(m, off, 32));

    float s = 0.0f;
    #pragma unroll
    for (int i = 0; i < 16; ++i) {
        v[i].x = expf(v[i].x - m);
        v[i].y = expf(v[i].y - m);
        v[i].z = expf(v[i].z - m);
        v[i].w = expf(v[i].w - m);
        s += (v[i].x + v[i].y) + (v[i].z + v[i].w);
    }
    #pragma unroll
    for (int off = 16; off > 0; off >>= 1)
        s += __shfl_xor(s, off, 32);

    const float inv = 1.0f / s;
    #pragma unroll
    for (int i = 0; i < 16; ++i) {
        v[i].x *= inv; v[i].y *= inv; v[i].z *= inv; v[i].w *= inv;
        *(float4*)&p[i * 128 + lane * 4] = v[i];
    }
}

// ===========================================================================
// Kernel 4/5: attn_out[b,s,h*64+d] = sum_k P[b,h,s,k] * V[b,h,k,d]
// Block tile 64(q) x 64(d); 8 waves = 4 M-subtiles x 2 N-halves,
// each wave 16x32 via 2 accumulators. K (keys) staged 16-wide in LDS.
// ===========================================================================
__global__ __launch_bounds__(256)
void pv_kernel(const float* __restrict__ attn, const float* __restrict__ qkv,
               float* __restrict__ attn_out) {
    __shared__ float Ps[64][20];
    __shared__ float Vs[16][64];

    const int tid  = threadIdx.x;
    const int lane = tid & 31;
    const int wave = tid >> 5;
    const int wm   = wave & 3;
    const int wn   = wave >> 2;
    const int m16  = lane & 15;
    const int kh   = (lane >> 4) << 1;

    const int bh = blockIdx.y;
    const int b  = bh >> 4;
    const int h  = bh & 15;
    const int qBase = blockIdx.x * 64;

    const float* P  = attn + (size_t)bh * SEQ * SEQ;
    const float* Vp = qkv + (size_t)b * SEQ * QKV3 + 2 * EMBED + (size_t)h * HDIM;

    v8f acc0 = {}, acc1 = {};

    const int pRow = tid >> 2;          // 0..63
    const int pCol = (tid & 3) << 2;
    const int vRow = tid >> 4;          // 0..15
    const int vCol = (tid & 15) << 2;   // 0..60

    for (int kb = 0; kb < SEQ; kb += 16) {
        __syncthreads();
        *(float4*)&Ps[pRow][pCol] =
            *(const float4*)&P[(size_t)(qBase + pRow) * SEQ + kb + pCol];
        *(float4*)&Vs[vRow][vCol] =
            *(const float4*)&Vp[(size_t)(kb + vRow) * QKV3 + vCol];
        __syncthreads();

        const int nb = wn * 32;
        #pragma unroll
        for (int k0 = 0; k0 < 16; k0 += 4) {
            v2f a;
            a.x = Ps[wm * 16 + m16][k0 + kh];
            a.y = Ps[wm * 16 + m16][k0 + kh + 1];
            v2f b0, b1;
            b0.x = Vs[k0 + kh    ][nb + m16];
            b0.y = Vs[k0 + kh + 1][nb + m16];
            b1.x = Vs[k0 + kh    ][nb + 16 + m16];
            b1.y = Vs[k0 + kh + 1][nb + 16 + m16];
            acc0 = wmma_f32(a, b0, acc0);
            acc1 = wmma_f32(a, b1, acc1);
        }
    }

    const int rowOff = (lane >> 4) * 8;
    const int c0 = h * HDIM + wn * 32 + m16;
    #pragma unroll
    for (int r = 0; r < 8; ++r) {
        const size_t row = ((size_t)b * SEQ + qBase + wm * 16 + rowOff + r) * EMBED;
        attn_out[row + c0     ] = acc0[r];
        attn_out[row + c0 + 16] = acc1[r];
    }
}

// ===========================================================================
// Launch: qkv-proj -> scores -> softmax -> P·V -> out-proj
// ===========================================================================
extern "C" void kernel_launch(void* const* d_in, const int* in_sizes, int n_in,
                              void* d_out, int out_size, void* d_ws, size_t ws_size,
                              hipStream_t stream) {
    const float* x     = (const float*)d_in[0];
    const float* qkv_w = (const float*)d_in[1];
    const float* qkv_b = (const float*)d_in[2];
    const float* out_w = (const float*)d_in[3];
    const float* out_b = (const float*)d_in[4];

    float* out  = (float*)d_out;                              // [B,S,E]
    float* attn = out + (size_t)BATCH * SEQ * EMBED;          // [B,H,S,S]

    float* qkv      = (float*)d_ws;                           // [B,S,3E] 48MB
    float* attn_out = qkv + (size_t)BATCH * SEQ * QKV3;       // [B,S,E]  16MB

    // 1) qkv = x @ qkv_w^T + qkv_b          M=4096, K=1024, F=3072
    gemm_nt_bias<<<dim3(BATCH * SEQ / 64, QKV3 / 128), 256, 0, stream>>>(
        x, qkv_w, qkv_b, qkv, EMBED, QKV3);

    // 2) scores = (Q K^T) / 8               per (b,h): 2048 x 2048, D=64
    scores_kernel<<<dim3(SEQ / 64, SEQ / 128, BATCH * NHEAD), 256, 0, stream>>>(
        qkv, attn);

    // 3) softmax rows (in place, writes attn_weights output)
    softmax_kernel<<<(BATCH * NHEAD * SEQ) / 8, 256, 0, stream>>>(attn);

    // 4) attn_out = P @ V                   per (b,h): 2048 x 64, K=2048
    pv_kernel<<<dim3(SEQ / 64, BATCH * NHEAD), 256, 0, stream>>>(
        attn, qkv, attn_out);

    // 5) output = attn_out @ out_w^T + out_b   M=4096, K=1024, F=1024
    gemm_nt_bias<<<dim3(BATCH * SEQ / 64, EMBED / 128), 256, 0, stream>>>(
        attn_out, out_w, out_b, out, EMBED, EMBED);
}